// Transformer_81372450390117
// MI455X (gfx1250) — compile-verified
//
#include <hip/hip_runtime.h>
#include <hip/hip_bf16.h>

typedef __attribute__((ext_vector_type(2))) float v2f;
typedef __attribute__((ext_vector_type(8))) float v8f;

#define B_ 8
#define T_ 2048
#define H_ 5
#define V_ 10

// ---------------------------------------------------------------------------
// Flash attention with WMMA f32 16x16x4 for Q·K^T.
// One wave = one 16-row query tile. Softmax state is kept per-lane (each lane
// owns one key column of the 16-wide tile across all 128 key tiles) so the
// inner loop has ZERO cross-lane operations; the 16 lane-partials are merged
// once at the end with the split-softmax combine rule.
// ---------------------------------------------------------------------------
__global__ __launch_bounds__(256) void attn_fa_wmma_kernel(
    const int* __restrict__ ids,
    const float* __restrict__ rope_offsets,
    const float* __restrict__ embed_const,
    const float* __restrict__ decode_eps,
    const float* __restrict__ qk_scale,
    float* __restrict__ att0)
{
    // kvs[s] = { x0[s]*cos(s), x0[s]*sin(s), digit[s], 0 }  (one ds_load_b128)
    __shared__ float4 kvs[T_];

    const int groupsPerBH = T_ / (16 * 8);      // 16
    const int grp = blockIdx.x % groupsPerBH;
    const int bh  = blockIdx.x / groupsPerBH;
    const int h = bh % H_;
    const int b = bh / H_;

    const float C    = embed_const[0];
    const float quad = decode_eps[0] * 0.5f;
    const float qk   = qk_scale[0];
    const float off  = rope_offsets[h];

    const int* tok = ids + b * T_;
    __builtin_prefetch(tok, 0, 1);              // -> global_prefetch_b8

    for (int s = threadIdx.x; s < T_; s += blockDim.x) {
        float d  = (float)tok[s];
        float x0 = C - quad * d * d;
        float sn, cs;
        __sincosf((float)s, &sn, &cs);
        kvs[s] = make_float4(x0 * cs, x0 * sn, d, 0.0f);
    }
    __syncthreads();

    const int lane = threadIdx.x & 31;
    const int wave = threadIdx.x >> 5;
    const int n    = lane & 15;   // key column within tile (owned by this lane)
    const int half = lane >> 4;   // C-layout: rows M<8 vs M>=8
    const int t0   = (grp * 8 + wave) * 16;

    // A operand: Q tile 16x4 f32 (K=2,3 zero -> lanes 16..31 zero).
    // qk_scale^2 / sqrt(head_dim) folded into Q so WMMA output = final score.
    v2f a = {0.0f, 0.0f};
    const float qfac = qk * qk * 0.70710678118654752440f;
    if (half == 0) {
        int   t  = t0 + n;
        float d  = (float)tok[t];
        float x0 = C - quad * d * d;
        float sn, cs;
        __sincosf((float)t - off, &sn, &cs);
        a.x = x0 * qfac * cs;
        a.y = x0 * qfac * sn;
    }

    // Per-row mask term for the diagonal tile (depends only on n vs row id),
    // precomputed so the steady-state loop carries no compares.
    float diagm[8];
#pragma unroll
    for (int r = 0; r < 8; ++r)
        diagm[r] = (n > r + 8 * half) ? -10000.0f : 0.0f;

    float m[8], l[8], acc[8];
#pragma unroll
    for (int r = 0; r < 8; ++r) { m[r] = -3.0e38f; l[r] = 0.0f; acc[r] = 0.0f; }

    // One key tile: WMMA scores + lane-local online softmax update.
    auto tile = [&](int s0, float uniformMask, bool diag) {
        const float4 kv = kvs[s0 + n];
        v2f bmat = {0.0f, 0.0f};
        if (half == 0) { bmat.x = kv.x; bmat.y = kv.y; }
        const float vval = kv.z;

        v8f c = {};
        c = __builtin_amdgcn_wmma_f32_16x16x4_f32(
                false, a, false, bmat, (short)0, c, false, false);

#pragma unroll
        for (int r = 0; r < 8; ++r) {
            const float sc = c[r] + (diag ? diagm[r] : uniformMask);
            const float mo = m[r];
            const float mx = fmaxf(sc, mo);
            const float mn = fminf(sc, mo);
            const float e  = __expf(mn - mx);   // single exp per row per tile
            const bool  up = sc > mo;           // new max came from this tile
            const float p    = up ? 1.0f : e;
            const float resc = up ? e : 1.0f;
            m[r]   = mx;
            l[r]   = fmaf(l[r],   resc, p);
            acc[r] = fmaf(acc[r], resc, p * vval);
        }
    };

    // Reference adds only -1e4 on the upper triangle (negligible vs ~5e10
    // scores => effectively dense softmax), so all T/16 key tiles are visited.
    for (int s0 = 0; s0 < t0; s0 += 16)        tile(s0, 0.0f,      false);
    tile(t0, 0.0f, true);
    for (int s0 = t0 + 16; s0 < T_; s0 += 16)  tile(s0, -10000.0f, false);

    // Merge the 16 lane-partials per row (once per wave): split-softmax rule.
#pragma unroll
    for (int r = 0; r < 8; ++r) {
        float mt = m[r];
#pragma unroll
        for (int msk = 1; msk < 16; msk <<= 1)
            mt = fmaxf(mt, __shfl_xor(mt, msk, 32));
        const float sc = __expf(m[r] - mt);
        float lw = l[r]   * sc;
        float aw = acc[r] * sc;
#pragma unroll
        for (int msk = 1; msk < 16; msk <<= 1) {
            lw += __shfl_xor(lw, msk, 32);
            aw += __shfl_xor(aw, msk, 32);
        }
        acc[r] = aw / lw;
    }

    if (n == 0) {
#pragma unroll
        for (int r = 0; r < 8; ++r) {
            const int t = t0 + r + 8 * half;
            att0[(b * H_ + h) * T_ + t] = acc[r];
        }
    }
}

// ---------------------------------------------------------------------------
// Epilogue: o_proj (only component 0 of each head is nonzero), residual,
// 4-wide MLP (W2 row 0 zero), output scaling, vocab projection (V=10).
// ---------------------------------------------------------------------------
__global__ __launch_bounds__(256) void epilogue_kernel(
    const int* __restrict__ ids,
    const float* __restrict__ o_w,
    const float* __restrict__ w1_abc,
    const float* __restrict__ w2_s,
    const float* __restrict__ embed_const,
    const float* __restrict__ decode_eps,
    const float* __restrict__ att0,
    float* __restrict__ out)
{
    const int idx = blockIdx.x * blockDim.x + threadIdx.x;  // b*T + t
    if (idx >= B_ * T_) return;
    const int b = idx / T_;
    const int t = idx - b * T_;

    const float C    = embed_const[0];
    const float eps  = decode_eps[0];
    const float quad = eps * 0.5f;

    const float d  = (float)ids[idx];
    float x0 = C - quad * d * d;
    float x1 = d;

    const float a0 = att0[(b * H_ + 0) * T_ + t];
    const float a1 = att0[(b * H_ + 1) * T_ + t];
    const float a2 = att0[(b * H_ + 2) * T_ + t];
    const float a3 = att0[(b * H_ + 3) * T_ + t];
    const float a4 = att0[(b * H_ + 4) * T_ + t];

    x0 += o_w[0] * a0 + o_w[1] * a1 + o_w[2] * a2;
    x1 += o_w[3] * a0 + o_w[4] * a3 + o_w[5] * a4;

    const float wa = w1_abc[0], wb = w1_abc[1], wc = w1_abc[2];
    const float h0 = fmaxf(wa * x0 + (C - 8.0f), 0.0f);
    const float h1 = fmaxf(wa * x0 + (C - 9.0f), 0.0f);
    const float h2 = fmaxf(wb * x0 + wc * x1 + (2.0f * C - 188.0f), 0.0f);
    const float h3 = fmaxf(wb * x0 + wc * x1 + (2.0f * C - 189.0f), 0.0f);

    const float s1 = w2_s[0], s10 = w2_s[1];
    x1 += s1 * h0 - s1 * h1 - s10 * h2 + s10 * h3;

    const float y0 = x0 / C;
    const float y1 = x1 * eps;

    float* o = out + idx * V_;
#pragma unroll
    for (int v = 0; v < V_; ++v) {
        const float fv = (float)v;
        o[v] = y0 * (C - quad * fv * fv) + y1 * fv;
    }
}

extern "C" void kernel_launch(void* const* d_in, const int* in_sizes, int n_in,
                              void* d_out, int out_size, void* d_ws, size_t ws_size,
                              hipStream_t stream) {
    (void)in_sizes; (void)n_in; (void)out_size; (void)ws_size;
    const int*   ids          = (const int*)  d_in[0];
    const float* o_w          = (const float*)d_in[1];
    const float* w1_abc       = (const float*)d_in[2];
    const float* w2_s         = (const float*)d_in[3];
    const float* embed_const  = (const float*)d_in[4];
    const float* decode_eps   = (const float*)d_in[5];
    const float* qk_scale     = (const float*)d_in[6];
    const float* rope_offsets = (const float*)d_in[7];

    float* att0 = (float*)d_ws;           // B*H*T floats = 320 KB scratch
    float* out  = (float*)d_out;          // B*T*V floats

    const int nBlocks = B_ * H_ * (T_ / 128);   // 640 blocks, 8 wave32s each
    attn_fa_wmma_kernel<<<nBlocks, 256, 0, stream>>>(
        ids, rope_offsets, embed_const, decode_eps, qk_scale, att0);

    const int n = B_ * T_;
    epilogue_kernel<<<(n + 255) / 256, 256, 0, stream>>>(
        ids, o_w, w1_abc, w2_s, embed_const, decode_eps, att0, out);
}